// LearnableMask_51427938402772
// MI455X (gfx1250) — compile-verified
//
#include <hip/hip_runtime.h>

#define D_MODEL  1024
#define N_LAYERS 24
#define KEEP     12
#define BLOCK    256

typedef __attribute__((ext_vector_type(2))) float v2f;
typedef __attribute__((ext_vector_type(4))) float v4f;
typedef __attribute__((ext_vector_type(8))) float v8f;
typedef __attribute__((ext_vector_type(4))) int   v4i;

// Pointer flavors for the async global->LDS DMA builtin (param 0 is a
// v4i pointer in the global/device address space per clang's diagnostic).
typedef __attribute__((address_space(1))) v4i* gv4i_p;
typedef __attribute__((address_space(3))) v4i* lv4i_p;

__global__ __launch_bounds__(BLOCK) void
lmask_fused_kernel(const float* __restrict__ x, const float* __restrict__ wg,
                   float* __restrict__ out) {
  __shared__ __align__(16) float xs[N_LAYERS * D_MODEL];  // 96 KB token tile
  __shared__ __align__(16) float wgs[D_MODEL];            // 4 KB gate weights
  __shared__ float scoresS[N_LAYERS];

  const int tid  = threadIdx.x;
  const int lane = tid & 31;
  const int wv   = tid >> 5;                       // wave id 0..7
  const long tok = blockIdx.x;
  const float* __restrict__ xa = x + tok * (long)(N_LAYERS * D_MODEL);

  // ---- stage w_g into LDS (coalesced) ----
#pragma unroll
  for (int j = 0; j < D_MODEL / BLOCK; ++j)
    wgs[tid + BLOCK * j] = wg[tid + BLOCK * j];

  // ---- stage 24x1024 f32 token tile into LDS ----
  // Prefer the CDNA5 async global->LDS DMA path; fall back to vec4 copies.
#if defined(__gfx1250__) && __has_builtin(__builtin_amdgcn_global_load_async_to_lds_b128)
  {
#pragma unroll
    for (int j = 0; j < (N_LAYERS * D_MODEL) / (4 * BLOCK); ++j) {
      const int idx = tid + BLOCK * j;  // vec4 index
      __builtin_amdgcn_global_load_async_to_lds_b128(
          (gv4i_p)(xa + 4 * idx),
          (lv4i_p)(xs + 4 * idx),
          0, 0);
    }
  }
#if __has_builtin(__builtin_amdgcn_s_wait_asynccnt)
  __builtin_amdgcn_s_wait_asynccnt(0);
#else
  asm volatile("s_wait_asynccnt 0x0" ::: "memory");
#endif
#else
  {
#pragma unroll
    for (int j = 0; j < (N_LAYERS * D_MODEL) / (4 * BLOCK); ++j) {
      const int idx = tid + BLOCK * j;
      ((v4f*)xs)[idx] = ((const v4f*)xa)[idx];
    }
  }
#endif
  __syncthreads();

  // ---- gate scores: 8 waves x 3 layers, lane-strided dot + shfl-xor reduce ----
#pragma unroll
  for (int s = 0; s < N_LAYERS / 8; ++s) {
    const int n = wv * (N_LAYERS / 8) + s;
    const float* row = xs + n * D_MODEL;
    float p = 0.0f;
#pragma unroll
    for (int i = 0; i < D_MODEL / 32; ++i)
      p = fmaf(row[lane + 32 * i], wgs[lane + 32 * i], p);
#pragma unroll
    for (int m = 16; m >= 1; m >>= 1)
      p += __shfl_xor(p, m, 32);
    if (lane == 0) scoresS[n] = p;
  }
  __syncthreads();

  // ---- top-12 keep mask (top_k tie-break: lower index wins) + masked softmax ----
  float sc[N_LAYERS];
#pragma unroll
  for (int n = 0; n < N_LAYERS; ++n) sc[n] = scoresS[n];

  float wgt[N_LAYERS];
  bool  kept[N_LAYERS];
  float mx = -__builtin_inff();
#pragma unroll
  for (int n = 0; n < N_LAYERS; ++n) {
    int rank = 0;
#pragma unroll
    for (int m = 0; m < N_LAYERS; ++m)
      rank += (sc[m] > sc[n]) || ((sc[m] == sc[n]) && (m < n));
    kept[n] = (rank < KEEP);
    if (kept[n] && sc[n] > mx) mx = sc[n];
  }
  float Z = 0.0f;
#pragma unroll
  for (int n = 0; n < N_LAYERS; ++n) {
    const float e = kept[n] ? __expf(sc[n] - mx) : 0.0f;
    wgt[n] = e;
    Z += e;
  }
  const float rz = 1.0f / Z;
#pragma unroll
  for (int n = 0; n < N_LAYERS; ++n) wgt[n] *= rz;

  // ---- combine: out(1x1024) = w(1x24) * X(24x1024) via V_WMMA_F32_16X16X4_F32 ----
  // A (16x4, row M=0 only): lane0 -> {K0,K1}, lane16 -> {K2,K3}, all else 0.
  // B (4x16): v0 = rows K0|K2 in lane halves, v1 = rows K1|K3 (mirrors A/C layout).
  // D row M=0 lives in VGPR0, lanes 0..15.
#if __has_builtin(__builtin_amdgcn_wmma_f32_16x16x4_f32)
  v2f A[N_LAYERS / 4];
#pragma unroll
  for (int kc = 0; kc < N_LAYERS / 4; ++kc) {
    float a0 = 0.0f, a1 = 0.0f;
    if (lane == 0)  { a0 = wgt[4 * kc + 0]; a1 = wgt[4 * kc + 1]; }
    if (lane == 16) { a0 = wgt[4 * kc + 2]; a1 = wgt[4 * kc + 3]; }
    A[kc][0] = a0;
    A[kc][1] = a1;
  }
  const int colInLane = lane & 15;
  const int rowSel    = (lane >> 4) << 1;  // 0 for lanes 0-15, 2 for lanes 16-31
  float* __restrict__ obase = out + tok * (long)D_MODEL;
#pragma unroll
  for (int t = 0; t < 8; ++t) {
    const int cb = (wv * 8 + t) * 16;  // 8 waves x 8 tiles x 16 cols = 1024
    v8f c = {0.f, 0.f, 0.f, 0.f, 0.f, 0.f, 0.f, 0.f};
#pragma unroll
    for (int kc = 0; kc < N_LAYERS / 4; ++kc) {
      const int k0 = 4 * kc + rowSel;
      v2f b;
      b[0] = xs[k0 * D_MODEL + cb + colInLane];
      b[1] = xs[(k0 + 1) * D_MODEL + cb + colInLane];
      c = __builtin_amdgcn_wmma_f32_16x16x4_f32(
              false, A[kc], false, b, (short)0, c, false, false);
    }
    if (lane < 16) obase[cb + lane] = c[0];
  }
#else
  // VALU fallback combine (4 columns per thread, coalesced)
  float* __restrict__ obase = out + tok * (long)D_MODEL;
#pragma unroll
  for (int j = 0; j < D_MODEL / BLOCK; ++j) {
    const int d = tid + BLOCK * j;
    float acc = 0.0f;
#pragma unroll
    for (int n = 0; n < N_LAYERS; ++n)
      acc = fmaf(wgt[n], xs[n * D_MODEL + d], acc);
    obase[d] = acc;
  }
#endif
}

extern "C" void kernel_launch(void* const* d_in, const int* in_sizes, int n_in,
                              void* d_out, int out_size, void* d_ws, size_t ws_size,
                              hipStream_t stream) {
  (void)n_in; (void)out_size; (void)d_ws; (void)ws_size;
  const float* x  = (const float*)d_in[0];
  const float* wg = (const float*)d_in[1];
  float* out = (float*)d_out;
  const int ntok = in_sizes[0] / (N_LAYERS * D_MODEL);  // B*T = 8000
  lmask_fused_kernel<<<ntok, BLOCK, 0, stream>>>(x, wg, out);
}